// TexualEmbeddingLayer_50843822850316
// MI455X (gfx1250) — compile-verified
//
#include <hip/hip_runtime.h>
#include <hip/hip_bf16.h>
#include <math.h>

// Problem constants (from reference)
#define BSZ   64
#define TLEN  511
#define SLEN  512
#define DIN   768
#define EDIM  1024
#define HDIM  512
#define KTOP  153
#define MROWS (BSZ * KTOP)   // 9792 = 306 * 32
#define PAD_ID 1

typedef __attribute__((ext_vector_type(2))) float v2f;
typedef __attribute__((ext_vector_type(8))) float v8f;

// ---------------------------------------------------------------------------
// Kernel 1: lengths / eos row / masked row / stable top-k ranks / pool_len
// One 512-thread block per batch element.
// ---------------------------------------------------------------------------
__global__ __launch_bounds__(512) void setup_topk(
    const int* __restrict__ text, const float* __restrict__ atten,
    int* __restrict__ idx, int* __restrict__ plen)
{
    __shared__ float srow[SLEN];
    __shared__ int   scnt;
    const int b = blockIdx.x;
    const int tid = threadIdx.x;

    if (tid == 0) scnt = 0;
    __syncthreads();
    if (tid < TLEN) {
        if (text[b * TLEN + tid] != PAD_ID) atomicAdd(&scnt, 1);
    }
    __syncthreads();
    const int len = scnt + 1;           // mask = [1 | text_mask], lengths
    const int eos = len - 1;

    float v = atten[(size_t)b * SLEN * SLEN + (size_t)eos * SLEN + tid];
    if (tid == 0 || tid == eos) v = -1.0f;
    const float mask = (tid == 0) ? 1.0f : (((tid - 1) < (len - 1)) ? 1.0f : 0.0f);
    v *= mask;
    srow[tid] = v;
    __syncthreads();

    // Stable descending rank (ties broken by lower index) == jax.lax.top_k order
    int r = 0;
    for (int j = 0; j < SLEN; ++j) {
        const float vj = srow[j];
        r += (vj > v) || (vj == v && j < tid);
    }
    if (r < KTOP) idx[b * KTOP + r] = tid;

    if (tid == 0) {
        int pl = len - 2;
        if (pl < 1) pl = 1;
        if (pl > KTOP) pl = KTOP;
        plen[b] = pl;
    }
}

// ---------------------------------------------------------------------------
// Kernel 2: gather selected feature rows + L2 normalize. One block per row.
// ---------------------------------------------------------------------------
__global__ __launch_bounds__(256) void gather_l2norm(
    const float* __restrict__ features, const int* __restrict__ idx,
    float* __restrict__ feats)
{
    const int r = blockIdx.x;           // 0 .. MROWS-1
    const int b = r / KTOP;
    const int src = idx[r];
    const float* fin = features + ((size_t)b * SLEN + src) * DIN;
    float* fout = feats + (size_t)r * DIN;
    const int tid = threadIdx.x;

    const float v0 = fin[tid];
    const float v1 = fin[tid + 256];
    const float v2 = fin[tid + 512];

    __shared__ float sp[256];
    sp[tid] = v0 * v0 + v1 * v1 + v2 * v2;
    __syncthreads();
    for (int s = 128; s > 0; s >>= 1) {
        if (tid < s) sp[tid] += sp[tid + s];
        __syncthreads();
    }
    const float inv = 1.0f / (sqrtf(sp[0]) + 1e-8f);
    fout[tid]       = v0 * inv;
    fout[tid + 256] = v1 * inv;
    fout[tid + 512] = v2 * inv;
}

// ---------------------------------------------------------------------------
// f32 GEMM via V_WMMA_F32_16X16X4_F32, one wave per 32(M) x 64(N) tile.
// C[M,N] = A[M,Kd] * W[Kd,N] + bias[N] (+ resid[M,N]).
// Per K-step of 4: two b64 A-fragment loads (rows tileM / tileM+16) and 8 B
// scalars (pointer-bumped, immediate offsets) feed 8 WMMAs -> 1.25 loads/WMMA.
//   A frag (16x4): lanes 0-15 -> M=l, regs = K {kk,kk+1}; lanes 16-31 -> K {kk+2,kk+3}
//   B frag (4x16): lanes 0-15 -> N=l, regs = K {kk,kk+1}; lanes 16-31 -> K {kk+2,kk+3}
//   C/D: VGPR i -> M = i (lanes 0-15) / i+8 (lanes 16-31), N = lane&15
// Guard is wave-uniform so EXEC is all-ones at every WMMA.
// ---------------------------------------------------------------------------
__global__ __launch_bounds__(256) void gemm_wmma_f32(
    const float* __restrict__ A, const float* __restrict__ W,
    const float* __restrict__ bias, const float* __restrict__ resid,
    float* __restrict__ C, int M, int N, int Kd)
{
    const int wave = blockIdx.x * (blockDim.x >> 5) + (threadIdx.x >> 5);
    const int tilesN = N >> 6;               // 64-wide N tiles
    const int numTiles = (M >> 5) * tilesN;  // 32-tall M tiles
    if (wave >= numTiles) return;

    const int tileM = (wave / tilesN) << 5;
    const int tileN = (wave % tilesN) << 6;
    const int lane = threadIdx.x & 31;
    const int half = lane >> 4;    // 0: K pair {0,1}; 1: K pair {2,3}
    const int l    = lane & 15;

    const float* ap0 = A + (size_t)(tileM + l) * Kd + 2 * half;
    const float* ap1 = ap0 + (size_t)16 * Kd;
    const float* bp  = W + (size_t)(2 * half) * N + tileN + l;
    const size_t bstep = (size_t)4 * N;

    v8f acc00 = {}, acc01 = {}, acc02 = {}, acc03 = {};
    v8f acc10 = {}, acc11 = {}, acc12 = {}, acc13 = {};
    for (int kk = 0; kk < Kd; kk += 4) {
        const v2f a0 = *(const v2f*)ap0;  ap0 += 4;   // 8B-aligned
        const v2f a1 = *(const v2f*)ap1;  ap1 += 4;
        v2f b0, b1, b2, b3;
        b0.x = bp[0];      b0.y = bp[N];
        b1.x = bp[16];     b1.y = bp[N + 16];
        b2.x = bp[32];     b2.y = bp[N + 32];
        b3.x = bp[48];     b3.y = bp[N + 48];
        bp += bstep;
        acc00 = __builtin_amdgcn_wmma_f32_16x16x4_f32(false, a0, false, b0, (short)0, acc00, false, false);
        acc01 = __builtin_amdgcn_wmma_f32_16x16x4_f32(false, a0, false, b1, (short)0, acc01, false, false);
        acc02 = __builtin_amdgcn_wmma_f32_16x16x4_f32(false, a0, false, b2, (short)0, acc02, false, false);
        acc03 = __builtin_amdgcn_wmma_f32_16x16x4_f32(false, a0, false, b3, (short)0, acc03, false, false);
        acc10 = __builtin_amdgcn_wmma_f32_16x16x4_f32(false, a1, false, b0, (short)0, acc10, false, false);
        acc11 = __builtin_amdgcn_wmma_f32_16x16x4_f32(false, a1, false, b1, (short)0, acc11, false, false);
        acc12 = __builtin_amdgcn_wmma_f32_16x16x4_f32(false, a1, false, b2, (short)0, acc12, false, false);
        acc13 = __builtin_amdgcn_wmma_f32_16x16x4_f32(false, a1, false, b3, (short)0, acc13, false, false);
    }

    const int col = tileN + l;
    const float bv0 = bias[col];
    const float bv1 = bias[col + 16];
    const float bv2 = bias[col + 32];
    const float bv3 = bias[col + 48];
    #pragma unroll
    for (int i = 0; i < 8; ++i) {
        const int row0 = tileM + i + (half << 3);
        const size_t base0 = (size_t)row0 * N + col;
        const size_t base1 = base0 + (size_t)16 * N;
        float v00 = acc00[i] + bv0, v01 = acc01[i] + bv1;
        float v02 = acc02[i] + bv2, v03 = acc03[i] + bv3;
        float v10 = acc10[i] + bv0, v11 = acc11[i] + bv1;
        float v12 = acc12[i] + bv2, v13 = acc13[i] + bv3;
        if (resid) {
            v00 += resid[base0];      v01 += resid[base0 + 16];
            v02 += resid[base0 + 32]; v03 += resid[base0 + 48];
            v10 += resid[base1];      v11 += resid[base1 + 16];
            v12 += resid[base1 + 32]; v13 += resid[base1 + 48];
        }
        C[base0]      = v00;  C[base0 + 16] = v01;
        C[base0 + 32] = v02;  C[base0 + 48] = v03;
        C[base1]      = v10;  C[base1 + 16] = v11;
        C[base1 + 32] = v12;  C[base1 + 48] = v13;
    }
}

// ---------------------------------------------------------------------------
// Kernel 4: BatchNorm training statistics (biased var).
// Each block owns 32 consecutive channels; every 32-lane row-group reads one
// coalesced 128B line per row, then LDS-reduce across the 8 row-groups.
// ---------------------------------------------------------------------------
__global__ __launch_bounds__(256) void bn_stats(
    const float* __restrict__ h, float* __restrict__ mu, float* __restrict__ var)
{
    const int tid  = threadIdx.x;
    const int c    = (blockIdx.x << 5) + (tid & 31);   // channel
    const int rgrp = tid >> 5;                          // 0..7
    float s = 0.0f, sq = 0.0f;
    for (int r = rgrp; r < MROWS; r += 8) {
        const float v = h[(size_t)r * HDIM + c];
        s += v; sq += v * v;
    }
    __shared__ float ss[256], ssq[256];
    ss[tid] = s; ssq[tid] = sq;
    __syncthreads();
    for (int st = 128; st >= 32; st >>= 1) {
        if (tid < st) { ss[tid] += ss[tid + st]; ssq[tid] += ssq[tid + st]; }
        __syncthreads();
    }
    if (tid < 32) {
        const float m = ss[tid] / (float)MROWS;
        mu[c]  = m;
        var[c] = ssq[tid] / (float)MROWS - m * m;
    }
}

// ---------------------------------------------------------------------------
// Kernel 5: BN normalize + affine + ReLU, in place.
// ---------------------------------------------------------------------------
__global__ __launch_bounds__(256) void bn_apply(
    float* __restrict__ h, const float* __restrict__ mu, const float* __restrict__ var,
    const float* __restrict__ gamma, const float* __restrict__ beta)
{
    const int i = blockIdx.x * 256 + threadIdx.x;     // MROWS*HDIM total
    const int c = i & (HDIM - 1);
    float v = (h[i] - mu[c]) * rsqrtf(var[c] + 1e-5f) * gamma[c] + beta[c];
    h[i] = v > 0.0f ? v : 0.0f;
}

// ---------------------------------------------------------------------------
// Kernel 7: masked max-pool over first pool_len[b] gathered rows.
// ---------------------------------------------------------------------------
__global__ __launch_bounds__(256) void pool_max(
    const float* __restrict__ outb, const int* __restrict__ plen,
    float* __restrict__ dout)
{
    const int b = blockIdx.x >> 2;                       // EDIM/256 = 4 blocks per batch
    const int e = ((blockIdx.x & 3) << 8) + threadIdx.x;
    const int pl = plen[b];
    const float* p = outb + (size_t)b * KTOP * EDIM + e;
    float m = -INFINITY;
    for (int k = 0; k < pl; ++k) m = fmaxf(m, p[(size_t)k * EDIM]);
    dout[b * EDIM + e] = m;
}

// ---------------------------------------------------------------------------
extern "C" void kernel_launch(void* const* d_in, const int* in_sizes, int n_in,
                              void* d_out, int out_size, void* d_ws, size_t ws_size,
                              hipStream_t stream)
{
    const float* features = (const float*)d_in[0];
    const int*   text     = (const int*)  d_in[1];
    const float* atten    = (const float*)d_in[2];
    const float* linear_w = (const float*)d_in[3];
    const float* linear_b = (const float*)d_in[4];
    const float* mlp_w0   = (const float*)d_in[5];
    const float* mlp_b0   = (const float*)d_in[6];
    const float* bn_gamma = (const float*)d_in[7];
    const float* bn_beta  = (const float*)d_in[8];
    const float* mlp_w1   = (const float*)d_in[9];
    const float* mlp_b1   = (const float*)d_in[10];
    float* out = (float*)d_out;

    // Workspace carve-up (~130 MB total)
    char* ws = (char*)d_ws;
    auto carve = [&](size_t bytes) -> char* {
        char* p = ws;
        ws += (bytes + 255) & ~(size_t)255;
        return p;
    };
    int*   idx   = (int*)  carve((size_t)MROWS * sizeof(int));
    int*   plen  = (int*)  carve((size_t)BSZ * sizeof(int));
    float* mu    = (float*)carve((size_t)HDIM * sizeof(float));
    float* var   = (float*)carve((size_t)HDIM * sizeof(float));
    float* feats = (float*)carve((size_t)MROWS * DIN  * sizeof(float));
    float* cap   = (float*)carve((size_t)MROWS * EDIM * sizeof(float));
    float* hbuf  = (float*)carve((size_t)MROWS * HDIM * sizeof(float));
    float* outb  = (float*)carve((size_t)MROWS * EDIM * sizeof(float));
    (void)ws_size; (void)in_sizes; (void)n_in; (void)out_size;

    // 1) lengths / eos row / top-k indices / pool lengths
    setup_topk<<<BSZ, 512, 0, stream>>>(text, atten, idx, plen);

    // 2) gather + l2norm
    gather_l2norm<<<MROWS, 256, 0, stream>>>(features, idx, feats);

    // 3) cap = feats @ linear_w + linear_b          [9792 x 1024, K=768]
    {
        const int waves = (MROWS / 32) * (EDIM / 64);   // 4896
        gemm_wmma_f32<<<(waves + 7) / 8, 256, 0, stream>>>(
            feats, linear_w, linear_b, nullptr, cap, MROWS, EDIM, DIN);
    }
    // 4) h = feats @ mlp_w0 + mlp_b0               [9792 x 512, K=768]
    {
        const int waves = (MROWS / 32) * (HDIM / 64);   // 2448
        gemm_wmma_f32<<<(waves + 7) / 8, 256, 0, stream>>>(
            feats, mlp_w0, mlp_b0, nullptr, hbuf, MROWS, HDIM, DIN);
    }
    // 5) BN stats + apply (+ReLU)
    bn_stats<<<HDIM / 32, 256, 0, stream>>>(hbuf, mu, var);
    bn_apply<<<(MROWS * HDIM) / 256, 256, 0, stream>>>(hbuf, mu, var, bn_gamma, bn_beta);

    // 6) out = h @ mlp_w1 + mlp_b1 + cap           [9792 x 1024, K=512]
    {
        const int waves = (MROWS / 32) * (EDIM / 64);   // 4896
        gemm_wmma_f32<<<(waves + 7) / 8, 256, 0, stream>>>(
            hbuf, mlp_w1, mlp_b1, cap, outb, MROWS, EDIM, HDIM);
    }

    // 7) masked max pool -> [64, 1024]
    pool_max<<<BSZ * (EDIM / 256), 256, 0, stream>>>(outb, plen, out);
}